// DINA_56968446214830
// MI455X (gfx1250) — compile-verified
//
#include <hip/hip_runtime.h>
#include <hip/hip_bf16.h>
#include <math.h>

// ---------------------------------------------------------------------------
// gfx1250 WMMA f32 16x16x4:  A = 16x4 (2 VGPRs), B = 4x16 (2 VGPRs),
// C/D = 16x16 f32 (8 VGPRs).  Codegen-verified in round 1.
// ---------------------------------------------------------------------------
typedef float v2f __attribute__((ext_vector_type(2)));
typedef float v8f __attribute__((ext_vector_type(8)));

__device__ __forceinline__ v8f wmma_f32x4(v2f a, v2f b, v8f c) {
    // (neg_a, A, neg_b, B, c_mod, C, reuse_a, reuse_b)
    return __builtin_amdgcn_wmma_f32_16x16x4_f32(false, a, false, b, (short)0, c,
                                                 false, false);
}

// Monotonic float <-> uint mapping for integer atomicMax/atomicMin on floats.
__device__ __forceinline__ unsigned fenc(float f) {
    unsigned u = __float_as_uint(f);
    return u ^ ((unsigned)((int)u >> 31) | 0x80000000u);
}
__device__ __forceinline__ float fdec(unsigned k) {
    unsigned u = (k & 0x80000000u) ? (k ^ 0x80000000u) : ~k;
    return __uint_as_float(u);
}

// Newton iteration for y^3/3 + y = x (the reference's nonsat_activation).
__device__ __forceinline__ float nonsat(float x) {
    float y = x;
#pragma unroll
    for (int t = 0; t < 12; ++t) {
        float y2 = y * y;
        y = (0.6666666666666667f * y2 * y + x) / (y2 + 1.0f);
    }
    return y;
}

#define NB   32
#define Q    1024
#define CDIM 256
#define LSEG 512

// ---------------------------------------------------------------------------
// Kernel 0: build M = concat(x1, pad(x2)) : (32, 1024, 256)
// ---------------------------------------------------------------------------
__global__ void dina_build_M(const float* __restrict__ x1,
                             const float* __restrict__ x2,
                             float* __restrict__ Mw) {
    size_t idx = (size_t)blockIdx.x * 256 + threadIdx.x; // 8,388,608 total
    int c = (int)(idx & 255);
    size_t r = idx >> 8;
    int i = (int)(r & 1023);
    int b = (int)(r >> 10);
    float v;
    if (i < LSEG) {
        v = x1[((size_t)b * LSEG + i) * CDIM + c];
    } else {
        int i2 = i - LSEG;
        v = (c < 192) ? x2[((size_t)b * LSEG + i2) * 192 + c] : 0.0f;
    }
    Mw[idx] = v;
}

// ---------------------------------------------------------------------------
// Kernel 1: init reduction buffers (per head; gmin only for head 0)
// ---------------------------------------------------------------------------
__global__ void dina_init_red(float* __restrict__ rmA, float* __restrict__ rmB,
                              unsigned* __restrict__ cmA, unsigned* __restrict__ cmB,
                              unsigned* __restrict__ gmin, int is_head0) {
    int idx = blockIdx.x * 256 + threadIdx.x;
    if (idx < NB * Q) {
        rmA[idx] = -INFINITY;
        rmB[idx] = -INFINITY;
        cmA[idx] = 0x007FFFFFu;   // fenc(-inf)
        cmB[idx] = 0x007FFFFFu;
    }
    if (idx == 0 && is_head0) *gmin = 0xFF800000u;  // fenc(+inf)
}

// ---------------------------------------------------------------------------
// Kernel 2: P[b] = M[b] @ U[head]   (1024x256 = 1024x256 * 256x256)
// One wave per 16x64 output block: 4 accumulators share each A load.
// ---------------------------------------------------------------------------
__global__ void dina_gemm1(const float* __restrict__ Mw,
                           const float* __restrict__ U,
                           float* __restrict__ P, int head) {
    int wave = (int)((blockIdx.x * blockDim.x + threadIdx.x) >> 5);
    int lane = (int)(threadIdx.x & 31);
    int b  = wave >> 8;            // 64 i-strips * 4 n-groups = 256 per batch
    int rr = wave & 255;
    int i0 = (rr >> 2) << 4;       // row strip (16 rows)
    int n0 = (rr & 3) << 6;        // 64-wide column group
    const float* A = Mw + (size_t)b * Q * CDIM;
    const float* B = U + (size_t)head * CDIM * CDIM;

    int m  = lane & 15;            // A-row / B-col within tile
    int kb = (lane >> 4) << 1;     // K pair base (0 or 2)

    v8f acc[4];
#pragma unroll
    for (int t = 0; t < 4; ++t) acc[t] = (v8f){0.f,0.f,0.f,0.f,0.f,0.f,0.f,0.f};

#pragma unroll 2
    for (int c0 = 0; c0 < CDIM; c0 += 4) {
        v2f a = *(const v2f*)(A + (size_t)(i0 + m) * CDIM + c0 + kb);
#pragma unroll
        for (int t = 0; t < 4; ++t) {
            v2f bv;
            bv.x = B[(size_t)(c0 + kb) * CDIM + n0 + 16 * t + m];
            bv.y = B[(size_t)(c0 + kb + 1) * CDIM + n0 + 16 * t + m];
            acc[t] = wmma_f32x4(a, bv, acc[t]);
        }
    }
    float* Pp = P + (size_t)b * Q * CDIM;
    int rowbase = i0 + ((lane >> 4) << 3);  // lanes 16..31 hold rows +8
#pragma unroll
    for (int t = 0; t < 4; ++t) {
        int col = n0 + 16 * t + m;
#pragma unroll
        for (int e = 0; e < 8; ++e)
            Pp[(size_t)(rowbase + e) * CDIM + col] = acc[t][e];
    }
}

// ---------------------------------------------------------------------------
// Kernel 3 (fused): K-strip = P[b][i0:i0+16,:] @ M[b]^T * 1/16 -> nonsat ->
// row-max (register, split by j-half), col-max (encoded atomicMax, split by
// i-half), global min (head 0). K is never materialized. 4 j-tiles per pass
// share every A load.
// ---------------------------------------------------------------------------
__global__ void dina_gemm2_fused(const float* __restrict__ Mw,
                                 const float* __restrict__ P,
                                 float* __restrict__ rmA, float* __restrict__ rmB,
                                 unsigned* __restrict__ cmA, unsigned* __restrict__ cmB,
                                 unsigned* __restrict__ gmin, int is_head0) {
    int wave = (int)((blockIdx.x * blockDim.x + threadIdx.x) >> 5);
    int lane = (int)(threadIdx.x & 31);
    int b  = wave >> 6;            // 64 row-strips per batch
    int i0 = (wave & 63) << 4;
    const float* A  = P  + (size_t)b * Q * CDIM;
    const float* Bm = Mw + (size_t)b * Q * CDIM;

    int m  = lane & 15;
    int kb = (lane >> 4) << 1;

    float rA[8], rB[8];
#pragma unroll
    for (int e = 0; e < 8; ++e) { rA[e] = -INFINITY; rB[e] = -INFINITY; }
    float lmin = INFINITY;

    unsigned* cmRow = (i0 < LSEG) ? cmA : cmB;  // i-half of this strip

#pragma unroll 1
    for (int j0 = 0; j0 < Q; j0 += 64) {
        // Hint the next 64-row B group toward the caches (global_prefetch_b8).
        if (j0 + 64 < Q) {
            __builtin_prefetch(Bm + (size_t)(j0 + 64 + lane) * CDIM, 0, 1);
            __builtin_prefetch(Bm + (size_t)(j0 + 96 + lane) * CDIM, 0, 1);
        }
        v8f acc[4];
#pragma unroll
        for (int t = 0; t < 4; ++t)
            acc[t] = (v8f){0.f,0.f,0.f,0.f,0.f,0.f,0.f,0.f};

#pragma unroll 2
        for (int c0 = 0; c0 < CDIM; c0 += 4) {
            // A: P rows; B[k][n] = M[j0+16t+n][c0+k] (contiguous float2).
            v2f a = *(const v2f*)(A + (size_t)(i0 + m) * CDIM + c0 + kb);
#pragma unroll
            for (int t = 0; t < 4; ++t) {
                v2f bv = *(const v2f*)(Bm + (size_t)(j0 + 16 * t + m) * CDIM + c0 + kb);
                acc[t] = wmma_f32x4(a, bv, acc[t]);
            }
        }
        // 64-aligned group => all 4 tiles in the same j-half.
        bool jleft = (j0 < LSEG);
#pragma unroll
        for (int t = 0; t < 4; ++t) {
            float cmax = -INFINITY;
#pragma unroll
            for (int e = 0; e < 8; ++e) {
                float x = acc[t][e] * 0.0625f;   // C^-0.5 = 1/16
                float v = nonsat(x);
                if (is_head0) lmin = fminf(lmin, v);
                if (jleft) rA[e] = fmaxf(rA[e], v);
                else       rB[e] = fmaxf(rB[e], v);
                cmax = fmaxf(cmax, v);
            }
            // column (j0+16t + lane&15) is shared by lane and lane^16
            cmax = fmaxf(cmax, __shfl_xor(cmax, 16, 32));
            if (lane < 16)
                atomicMax(&cmRow[(size_t)b * Q + j0 + 16 * t + lane], fenc(cmax));
        }
    }

    // row-max: lanes 0..15 share rows i0..i0+7, lanes 16..31 share i0+8..i0+15
#pragma unroll
    for (int e = 0; e < 8; ++e) {
        rA[e] = fmaxf(rA[e], __shfl_xor(rA[e], 1, 32));
        rA[e] = fmaxf(rA[e], __shfl_xor(rA[e], 2, 32));
        rA[e] = fmaxf(rA[e], __shfl_xor(rA[e], 4, 32));
        rA[e] = fmaxf(rA[e], __shfl_xor(rA[e], 8, 32));
        rB[e] = fmaxf(rB[e], __shfl_xor(rB[e], 1, 32));
        rB[e] = fmaxf(rB[e], __shfl_xor(rB[e], 2, 32));
        rB[e] = fmaxf(rB[e], __shfl_xor(rB[e], 4, 32));
        rB[e] = fmaxf(rB[e], __shfl_xor(rB[e], 8, 32));
    }
    if (lane == 0 || lane == 16) {
        int rowbase = i0 + ((lane >> 4) << 3);
#pragma unroll
        for (int e = 0; e < 8; ++e) {
            rmA[(size_t)b * Q + rowbase + e] = rA[e];
            rmB[(size_t)b * Q + rowbase + e] = rB[e];
        }
    }
    if (is_head0) {
        lmin = fminf(lmin, __shfl_xor(lmin, 1, 32));
        lmin = fminf(lmin, __shfl_xor(lmin, 2, 32));
        lmin = fminf(lmin, __shfl_xor(lmin, 4, 32));
        lmin = fminf(lmin, __shfl_xor(lmin, 8, 32));
        lmin = fminf(lmin, __shfl_xor(lmin, 16, 32));
        if (lane == 0) atomicMin(gmin, fenc(lmin));
    }
}

// ---------------------------------------------------------------------------
// Kernel 4: alpha = rows + cols (mask algebra), segment softmax, weighted sum
// against context -> out[head][b][seg][c]. One block per (b, seg).
// ---------------------------------------------------------------------------
__global__ void dina_finalize(const float* __restrict__ Mw,
                              const float* __restrict__ rmA, const float* __restrict__ rmB,
                              const unsigned* __restrict__ cmA, const unsigned* __restrict__ cmB,
                              const unsigned* __restrict__ gmin,
                              float* __restrict__ out, int head) {
    __shared__ float wls[LSEG];
    __shared__ float red[256];
    int tid = threadIdx.x;
    int b = blockIdx.x >> 1;
    int seg = blockIdx.x & 1;
    float minK = fdec(*gmin);

    float a01[2];
#pragma unroll
    for (int t = 0; t < 2; ++t) {
        int p = seg * LSEG + tid + t * 256;   // global position in [0,1024)
        size_t o = (size_t)b * Q + p;
        float rows = (p < LSEG) ? fmaxf(minK, rmB[o]) : fmaxf(rmA[o], rmB[o]);
        float cols = (p < LSEG) ? fmaxf(minK, fdec(cmB[o]))
                                : fmaxf(fdec(cmA[o]), fdec(cmB[o]));
        a01[t] = rows + cols;
    }
    wls[tid] = a01[0];
    wls[tid + 256] = a01[1];
    red[tid] = fmaxf(a01[0], a01[1]);
    __syncthreads();
    for (int s = 128; s > 0; s >>= 1) {
        if (tid < s) red[tid] = fmaxf(red[tid], red[tid + s]);
        __syncthreads();
    }
    float amax = red[0];
    __syncthreads();
    float e0 = expf(a01[0] - amax);
    float e1 = expf(a01[1] - amax);
    wls[tid] = e0;
    wls[tid + 256] = e1;
    red[tid] = e0 + e1;
    __syncthreads();
    for (int s = 128; s > 0; s >>= 1) {
        if (tid < s) red[tid] += red[tid + s];
        __syncthreads();
    }
    float inv = 1.0f / red[0];

    const float* Cx = Mw + ((size_t)b * Q + (size_t)seg * LSEG) * CDIM;
    float acc = 0.0f;
    for (int l = 0; l < LSEG; ++l)
        acc += wls[l] * Cx[(size_t)l * CDIM + tid];   // coalesced over tid
    acc *= inv;
    out[((((size_t)head * NB) + b) * 2 + seg) * CDIM + tid] = acc;
}

// ---------------------------------------------------------------------------
// Host-side launcher
// ---------------------------------------------------------------------------
extern "C" void kernel_launch(void* const* d_in, const int* in_sizes, int n_in,
                              void* d_out, int out_size, void* d_ws, size_t ws_size,
                              hipStream_t stream) {
    const float* x1 = (const float*)d_in[0];   // (32, 512, 256)
    const float* x2 = (const float*)d_in[1];   // (32, 512, 192)
    const float* U  = (const float*)d_in[2];   // (2, 256, 256)
    float* out = (float*)d_out;                // (2, 32, 2, 256)

    // Workspace layout
    float* Mw = (float*)d_ws;                          // 32*1024*256
    float* P  = Mw + (size_t)NB * Q * CDIM;            // 32*1024*256
    float* rmA = P + (size_t)NB * Q * CDIM;            // 32*1024
    float* rmB = rmA + NB * Q;
    unsigned* cmA = (unsigned*)(rmB + NB * Q);         // 32*1024
    unsigned* cmB = cmA + NB * Q;
    unsigned* gmin = cmB + NB * Q;                     // 1

    dina_build_M<<<(NB * Q * CDIM) / 256, 256, 0, stream>>>(x1, x2, Mw);

    for (int h = 0; h < 2; ++h) {
        dina_init_red<<<(NB * Q + 255) / 256, 256, 0, stream>>>(
            rmA, rmB, cmA, cmB, gmin, h == 0 ? 1 : 0);
        // 8192 waves (b * 64 strips * 4 n-groups), 8 waves/block -> 1024 blocks
        dina_gemm1<<<1024, 256, 0, stream>>>(Mw, U, P, h);
        // 2048 row-strips (b * 64), 8 waves/block -> 256 blocks
        dina_gemm2_fused<<<256, 256, 0, stream>>>(Mw, P, rmA, rmB, cmA, cmB,
                                                  gmin, h == 0 ? 1 : 0);
        dina_finalize<<<NB * 2, 256, 0, stream>>>(Mw, rmA, rmB, cmA, cmB, gmin,
                                                  out, h);
    }
}